// LagrangianConv3D_38878043963411
// MI455X (gfx1250) — compile-verified
//
#include <hip/hip_runtime.h>
#include <hip/hip_bf16.h>

typedef __attribute__((ext_vector_type(2))) float v2f;
typedef __attribute__((ext_vector_type(8))) float v8f;

#define GIN 128
#define GOUT 64

// ---------------------------------------------------------------------------
// out[n][j] = bias[0][j] + bias[1][j] + bias[2][j]
// ---------------------------------------------------------------------------
__global__ void init_out_kernel(float* __restrict__ out,
                                const float* __restrict__ bias, int total) {
    int i = blockIdx.x * blockDim.x + threadIdx.x;
    if (i < total) {
        int j = i & (GOUT - 1);
        out[i] = bias[j] + bias[GOUT + j] + bias[2 * GOUT + j];
    }
}

// ---------------------------------------------------------------------------
// support[k][n][0:64] = input[n][0:128] @ weight[k]   (f32 WMMA 16x16x4)
// 8 waves / block, one 16-row stripe per wave; N % 16 == 0 (tail waves clamp,
// duplicate identical-value stores are benign).
//
// A fragment (32-bit A 16x4): lanes 0-15 -> M=lane, v0=K0/v1=K1;
// lanes 16-31 -> M=lane-16, v0=K2/v1=K3  => per-lane base offset 2*half.
// B fragment mirrors: v0 = B[4kk+2*half][n], v1 = B[4kk+1+2*half][n].
// Weight is staged into LDS TRANSPOSED as lwT[kk][n][4] (the 4 K-values of a
// K-step contiguous per column), so a B fragment is ONE aligned ds_load_b64
// landing directly in an even VGPR pair (no packing movs).
// C/D layout: VGPR r -> row r (lanes 0-15) / r+8 (lanes 16-31), col = lane&15.
// ---------------------------------------------------------------------------
__global__ __launch_bounds__(256) void gemm_support_kernel(
    const float* __restrict__ input, const float* __restrict__ weight,
    float* __restrict__ support, int n_nodes) {
    __shared__ float lwT[GIN * GOUT];  // 32 KB: one k's weight, [kk][n][4]

    const int tid  = threadIdx.x;
    const int wave = tid >> 5;
    const int lane = tid & 31;
    const int half = lane >> 4;
    const int m    = lane & 15;

    int row0 = (blockIdx.x * 8 + wave) * 16;
    if (row0 + 16 > n_nodes) row0 = n_nodes - 16;

    // Cache this wave's 16x128 A-tile in registers in WMMA A-fragment layout.
    const float* ap = input + (size_t)(row0 + m) * GIN + 2 * half;
    v2f af[32];
#pragma unroll
    for (int kk = 0; kk < 32; ++kk)
        af[kk] = *(const v2f*)(ap + kk * 4);

    for (int k = 0; k < 3; ++k) {
        // Stage weight[k] (128x64 f32) into LDS, transposed to [kk][n][4].
        const float* wk = weight + (size_t)k * GIN * GOUT;
        for (int idx = tid; idx < GIN * GOUT / 4; idx += 256) {
            const int K  = idx >> 4;         // 0..127
            const int n0 = (idx & 15) * 4;   // 0..60
            float4 w = *(const float4*)&wk[K * GOUT + n0];
            const int kk = K >> 2, kq = K & 3;
            float* dst = &lwT[(kk * GOUT + n0) * 4 + kq];
            dst[0]  = w.x;   // n0+0
            dst[4]  = w.y;   // n0+1
            dst[8]  = w.z;   // n0+2
            dst[12] = w.w;   // n0+3
        }
        __syncthreads();

        float* sk = support + (size_t)k * n_nodes * GOUT;
        for (int nt = 0; nt < 4; ++nt) {
            const int n0 = nt * 16;
            const float* bbase = &lwT[(n0 + m) * 4 + 2 * half];
            v8f acc = {};
#pragma unroll
            for (int kk = 0; kk < 32; ++kk) {
                v2f b = *(const v2f*)(bbase + kk * GOUT * 4);  // ds_load_b64
                acc = __builtin_amdgcn_wmma_f32_16x16x4_f32(
                    false, af[kk], false, b, (short)0, acc, false, false);
            }
#pragma unroll
            for (int r = 0; r < 8; ++r)
                sk[(size_t)(row0 + r + 8 * half) * GOUT + n0 + m] = acc[r];
        }
        __syncthreads();
    }
}

// ---------------------------------------------------------------------------
// out[row[e]] += val[e] * support_k[col[e]]  ; 16 lanes per edge, float4/lane.
// Half-wave = one contiguous 256B gather; scatter via hardware f32 atomics.
// Edge lists are streamed once -> non-temporal loads keep L2 (192MB) free for
// the reused support (76.8MB) + out (25.6MB) working set.
// ---------------------------------------------------------------------------
__global__ __launch_bounds__(256) void spmm_scatter_kernel(
    const int* __restrict__ row, const int* __restrict__ col,
    const float* __restrict__ val, const float* __restrict__ support,
    float* __restrict__ out, int nedges) {
    const int tid = threadIdx.x;
    const int g   = tid >> 4;          // edge slot within block (0..15)
    const int q   = tid & 15;          // float4 slot within row  (0..15)
    const int e   = blockIdx.x * 16 + g;
    if (e >= nedges) return;

    const int   r = __builtin_nontemporal_load(row + e);
    const int   c = __builtin_nontemporal_load(col + e);
    const float v = __builtin_nontemporal_load(val + e);

    float4 s = *(const float4*)&support[(size_t)c * GOUT + q * 4];
    float* o = &out[(size_t)r * GOUT + q * 4];
    unsafeAtomicAdd(o + 0, v * s.x);
    unsafeAtomicAdd(o + 1, v * s.y);
    unsafeAtomicAdd(o + 2, v * s.z);
    unsafeAtomicAdd(o + 3, v * s.w);
}

// ---------------------------------------------------------------------------
extern "C" void kernel_launch(void* const* d_in, const int* in_sizes, int n_in,
                              void* d_out, int out_size, void* d_ws, size_t ws_size,
                              hipStream_t stream) {
    const float* input  = (const float*)d_in[0];
    const float* weight = (const float*)d_in[1];
    const float* bias   = (const float*)d_in[2];
    const int*   rx = (const int*)d_in[3];
    const int*   cx = (const int*)d_in[4];
    const float* vx = (const float*)d_in[5];
    const int*   ry = (const int*)d_in[6];
    const int*   cy = (const int*)d_in[7];
    const float* vy = (const float*)d_in[8];
    const int*   rz = (const int*)d_in[9];
    const int*   cz = (const int*)d_in[10];
    const float* vz = (const float*)d_in[11];

    const int n = in_sizes[0] / GIN;   // 100000
    const int e = in_sizes[3];         // 1600000

    float* out     = (float*)d_out;
    float* support = (float*)d_ws;     // 3 * n * 64 floats = 76.8 MB

    // 1) out = sum of biases
    init_out_kernel<<<(n * GOUT + 255) / 256, 256, 0, stream>>>(out, bias, n * GOUT);

    // 2) support[k] = input @ weight[k]
    const int row_tiles   = n / 16;                  // 6250
    const int gemm_blocks = (row_tiles + 7) / 8;     // 782
    gemm_support_kernel<<<gemm_blocks, 256, 0, stream>>>(input, weight, support, n);

    // 3) scatter-add each adjacency
    const int spmm_blocks = (e + 15) / 16;           // 100000
    spmm_scatter_kernel<<<spmm_blocks, 256, 0, stream>>>(
        rx, cx, vx, support + 0 * (size_t)n * GOUT, out, e);
    spmm_scatter_kernel<<<spmm_blocks, 256, 0, stream>>>(
        ry, cy, vy, support + 1 * (size_t)n * GOUT, out, e);
    spmm_scatter_kernel<<<spmm_blocks, 256, 0, stream>>>(
        rz, cz, vz, support + 2 * (size_t)n * GOUT, out, e);
}